// Yolo_Loss_37778532335941
// MI455X (gfx1250) — compile-verified
//
#include <hip/hip_runtime.h>
#include <hip/hip_bf16.h>

// Problem constants (from reference): S=56, B=5, C=20, N=128
#define NCDIM   20            // C
#define CP5     25            // C+5
#define PB      125           // B*(C+5) floats per cell in predicts
#define NANCH   5
#define NCELLS  (128*56*56)   // 401408
#define TILE    64            // cells per block
#define BLK     64            // threads per block (one cell per thread)
#define NTILES  (NCELLS/TILE) // 6272 (exact)

// ---------------- CDNA5 async global->LDS staging ----------------
#if defined(__gfx1250__) && __has_builtin(__builtin_amdgcn_global_load_async_to_lds_b128)
#define ASYNC_LDS 1
#else
#define ASYNC_LDS 0
#endif

typedef int v4i __attribute__((vector_size(16)));
typedef __attribute__((address_space(1))) v4i glb_v4i;   // global (AS1)
typedef __attribute__((address_space(3))) v4i lds_v4i;   // LDS (AS3)

__device__ __forceinline__ void copy16(const float* __restrict__ g, float* l) {
#if ASYNC_LDS
    // GLOBAL_LOAD_ASYNC_TO_LDS_B128: 16 bytes per lane, mem -> LDS, ASYNCcnt-tracked
    __builtin_amdgcn_global_load_async_to_lds_b128(
        (glb_v4i*)(void*)(const_cast<float*>(g)), (lds_v4i*)l, /*offset=*/0, /*cpol=*/0);
#else
    const float4 v = *(const float4*)g;
    *(float4*)l = v;
#endif
}

__device__ __forceinline__ void wait_stage_and_barrier() {
#if ASYNC_LDS
#if __has_builtin(__builtin_amdgcn_s_wait_asynccnt)
    __builtin_amdgcn_s_wait_asynccnt(0);
#else
    asm volatile("s_wait_asynccnt 0" ::: "memory");
#endif
#endif
    __syncthreads();
}

// ---------------- main per-cell loss kernel ----------------
__global__ __launch_bounds__(BLK) void yolo_loss_kernel(
    const float* __restrict__ predicts,   // [NCELLS * 125]
    const float* __restrict__ targets,    // [NCELLS * 25]
    const float* __restrict__ anchors,    // [5 * 2]
    float* __restrict__ partials)         // [NTILES]
{
    __shared__ float ldsP[TILE * PB];     // 32000 B
    __shared__ float ldsT[TILE * CP5];    //  6400 B
    __shared__ float red[BLK];

    const int tid = threadIdx.x;
    const int tileBase = blockIdx.x * TILE;
    const float* gP = predicts + (size_t)tileBase * PB;
    const float* gT = targets  + (size_t)tileBase * CP5;

    // Stage predicts tile: TILE*PB/4 = 2000 16B chunks = 31*64 + 16
#pragma unroll
    for (int k = 0; k < 31; ++k) {
        const int c = tid + k * BLK;
        copy16(gP + 4 * c, &ldsP[4 * c]);
    }
    if (tid < 16) { const int c = 1984 + tid; copy16(gP + 4 * c, &ldsP[4 * c]); }

    // Stage targets tile: TILE*CP5/4 = 400 chunks = 6*64 + 16
#pragma unroll
    for (int k = 0; k < 6; ++k) {
        const int c = tid + k * BLK;
        copy16(gT + 4 * c, &ldsT[4 * c]);
    }
    if (tid < 16) { const int c = 384 + tid; copy16(gT + 4 * c, &ldsT[4 * c]); }

    wait_stage_and_barrier();

    // Conflict-free LDS reads: strides 125 / 25 floats are odd -> distinct banks per lane.
    const float* tp = &ldsT[tid * CP5];
    const float* pp = &ldsP[tid * PB];

    // Target box (cxcywh -> xyxy), mirroring reference op order.
    const float tcx = tp[NCDIM + 1], tcy = tp[NCDIM + 2];
    const float tw  = tp[NCDIM + 3], th_ = tp[NCDIM + 4];
    const float tb0 = tcx - 0.5f * tw,  tb1 = tcy - 0.5f * th_;
    const float tb2 = tcx + 0.5f * tw,  tb3 = tcy + 0.5f * th_;
    const float area_t = (tb2 - tb0) * (tb3 - tb1);

    int   bi   = 0;
    float best = -3.0e38f;
#pragma unroll
    for (int b = 0; b < NANCH; ++b) {
        const float* p = pp + b * CP5;
        const float px = 1.0f / (1.0f + expf(-p[NCDIM + 1]));
        const float py = 1.0f / (1.0f + expf(-p[NCDIM + 2]));
        const float pw = expf(p[NCDIM + 3]) * anchors[2 * b];
        const float ph = expf(p[NCDIM + 4]) * anchors[2 * b + 1];
        const float pb0 = px - 0.5f * pw, pb1 = py - 0.5f * ph;
        const float pb2 = px + 0.5f * pw, pb3 = py + 0.5f * ph;
        const float lt0 = fmaxf(pb0, tb0), lt1 = fmaxf(pb1, tb1);
        const float rb0 = fminf(pb2, tb2), rb1 = fminf(pb3, tb3);
        const float wi = fmaxf(rb0 - lt0, 0.0f);
        const float hi = fmaxf(rb1 - lt1, 0.0f);
        const float inter  = wi * hi;
        const float area_p = (pb2 - pb0) * (pb3 - pb1);
        const float iou = inter / (area_p + area_t - inter);
        if (iou > best) { best = iou; bi = b; }   // strict > == argmax first-max
    }

    // Best-anchor prediction (decoded), pbest[0] *= best_idx
    const float* p  = pp + bi * CP5;
    const float bx  = 1.0f / (1.0f + expf(-p[NCDIM + 1]));
    const float by  = 1.0f / (1.0f + expf(-p[NCDIM + 2]));
    const float bw  = expf(p[NCDIM + 3]) * anchors[2 * bi];
    const float bh  = expf(p[NCDIM + 4]) * anchors[2 * bi + 1];
    const float ex  = tp[NCDIM];
    const float no  = 1.0f - ex;

    float loss = 0.0f, d;
    d = ex * (p[0] * (float)bi) - ex * tp[0]; loss += d * d;      // class 0 (scaled by idx)
#pragma unroll
    for (int j = 1; j < NCDIM; ++j) { d = ex * p[j] - ex * tp[j]; loss += d * d; }
    d = ex * p[NCDIM] - ex * tp[NCDIM]; loss += d * d;            // obj
    d = no * p[NCDIM] - no * tp[NCDIM]; loss += d * d;            // noobj
    d = ex * bx - ex * tp[NCDIM + 1];   loss += d * d;            // xc
    d = ex * by - ex * tp[NCDIM + 2];   loss += d * d;            // yc
    const float sgw = (bw > 0.0f) ? 1.0f : ((bw < 0.0f) ? -1.0f : 0.0f);
    const float sgh = (bh > 0.0f) ? 1.0f : ((bh < 0.0f) ? -1.0f : 0.0f);
    const float sw  = sgw * sqrtf(fabsf(bw) + 1e-6f);
    const float sh  = sgh * sqrtf(fabsf(bh) + 1e-6f);
    d = ex * sw - ex * tp[NCDIM + 3];   loss += d * d;            // w
    d = ex * sh - ex * tp[NCDIM + 4];   loss += d * d;            // h

    // Deterministic block tree-reduction
    red[tid] = loss;
    __syncthreads();
#pragma unroll
    for (int off = BLK / 2; off > 0; off >>= 1) {
        if (tid < off) red[tid] += red[tid + off];
        __syncthreads();
    }
    if (tid == 0) partials[blockIdx.x] = red[0];
}

// ---------------- deterministic final reduction ----------------
__global__ __launch_bounds__(256) void yolo_reduce_kernel(
    const float* __restrict__ partials, float* __restrict__ out, int n)
{
    __shared__ float red[256];
    float s = 0.0f;
    for (int i = threadIdx.x; i < n; i += 256) s += partials[i];
    red[threadIdx.x] = s;
    __syncthreads();
#pragma unroll
    for (int off = 128; off > 0; off >>= 1) {
        if (threadIdx.x < off) red[threadIdx.x] += red[threadIdx.x + off];
        __syncthreads();
    }
    if (threadIdx.x == 0) out[0] = red[0];
}

extern "C" void kernel_launch(void* const* d_in, const int* in_sizes, int n_in,
                              void* d_out, int out_size, void* d_ws, size_t ws_size,
                              hipStream_t stream) {
    const float* predicts = (const float*)d_in[0];
    const float* targets  = (const float*)d_in[1];
    const float* anchors  = (const float*)d_in[2];
    float* out      = (float*)d_out;
    float* partials = (float*)d_ws;   // NTILES * 4 bytes ≈ 25 KB

    yolo_loss_kernel<<<NTILES, BLK, 0, stream>>>(predicts, targets, anchors, partials);
    yolo_reduce_kernel<<<1, 256, 0, stream>>>(partials, out, NTILES);
}